// GNNEncoder_15496242004448
// MI455X (gfx1250) — compile-verified
//
#include <hip/hip_runtime.h>

// ---------------- problem constants (match reference) ----------------
namespace {
constexpr int   kN     = 100000;
constexpr int   kE     = 1600000;
constexpr int   kInC   = 256;
constexpr int   kHid   = 128;
constexpr int   kOutC  = 128;
constexpr int   kHeads = 4;
constexpr int   kCHead = 32;
constexpr float kBnEps = 1e-5f;
constexpr float kSlope = 0.2f;
}

typedef float v2f __attribute__((ext_vector_type(2)));
typedef float v8f __attribute__((ext_vector_type(8)));

// Monotone order-preserving encode of f32 into u32 for atomicMax-based segment max.
__device__ __forceinline__ unsigned int ord_enc(float f) {
  unsigned int i = __float_as_uint(f);
  return (i & 0x80000000u) ? ~i : (i | 0x80000000u);
}
__device__ __forceinline__ float ord_dec(unsigned int e) {
  unsigned int i = (e & 0x80000000u) ? (e & 0x7fffffffu) : ~e;
  return __uint_as_float(i);
}

// ---------------- fp32 WMMA GEMM: C[M x Nc] = A[M x K] * B[K x Nc] ----------------
// Nc must be 128. One wave per 16-row block; 8 accumulator tiles (16x16) cover 128 cols.
// V_WMMA_F32_16X16X4_F32 layouts (ISA 7.12.2):
//   A 16x4:  lanes 0-15 M=0..15 {v0:K=k, v1:K=k+1}; lanes 16-31 {v0:K=k+2, v1:K=k+3}
//   B 4x16:  lanes 0-15 N=0..15 {v0:K=k, v1:K=k+1}; lanes 16-31 {v0:K=k+2, v1:K=k+3}
//   C/D:     VGPR r: lanes 0-15 -> (M=r, N=lane); lanes 16-31 -> (M=r+8, N=lane-16)
__global__ void __launch_bounds__(128)
gemm_f32_wmma(const float* __restrict__ A, const float* __restrict__ B,
              float* __restrict__ C, int M, int K, int Nc) {
  const int lane = threadIdx.x & 31;
  const int wave = threadIdx.x >> 5;
  const int row0 = (blockIdx.x * 4 + wave) * 16;
  if (row0 >= M) return;                    // uniform per wave (EXEC stays all-1)
  const int half = lane >> 4;               // 0: lanes 0-15, 1: lanes 16-31
  const int l15  = lane & 15;

  const v8f vzero = {0.f, 0.f, 0.f, 0.f, 0.f, 0.f, 0.f, 0.f};
  v8f acc[8];
#pragma unroll
  for (int t = 0; t < 8; ++t) acc[t] = vzero;

  const float* arow = A + (size_t)(row0 + l15) * K;
  for (int k = 0; k < K; k += 4) {
    const int ka = k + half * 2;            // even -> 8B aligned
    v2f a = *(const v2f*)(arow + ka);
#pragma unroll
    for (int t = 0; t < 8; ++t) {
      const int col = t * 16 + l15;
      v2f b;
      b.x = B[(size_t)ka * Nc + col];
      b.y = B[(size_t)(ka + 1) * Nc + col];
      acc[t] = __builtin_amdgcn_wmma_f32_16x16x4_f32(
          /*neg_a=*/false, a, /*neg_b=*/false, b,
          /*c_mod=*/(short)0, acc[t], /*reuse_a=*/false, /*reuse_b=*/false);
    }
  }
#pragma unroll
  for (int t = 0; t < 8; ++t) {
    const int col = t * 16 + l15;
#pragma unroll
    for (int r = 0; r < 8; ++r) {
      const int row = row0 + r + half * 8;
      C[(size_t)row * Nc + col] = acc[t][r];
    }
  }
}

// ---------------- attention scores: a_src[n,h], a_dst[n,h] ----------------
__global__ void att_scores(const float* __restrict__ h,
                           const float* __restrict__ att_src,
                           const float* __restrict__ att_dst,
                           float* __restrict__ a_src, float* __restrict__ a_dst) {
  int t = blockIdx.x * blockDim.x + threadIdx.x;     // over N*HEADS
  if (t >= kN * kHeads) return;
  int n = t >> 2, hd = t & 3;
  const float* hp = h + (size_t)n * kHid + hd * kCHead;
  const float* as = att_src + hd * kCHead;
  const float* ad = att_dst + hd * kCHead;
  float s = 0.f, d = 0.f;
#pragma unroll 8
  for (int c = 0; c < kCHead; ++c) { float v = hp[c]; s += v * as[c]; d += v * ad[c]; }
  a_src[t] = s;
  a_dst[t] = d;
}

// ---------------- edge pass 1: e = leaky_relu(a_src[src]+a_dst[dst]); segment max ----------------
__global__ void edge_max(const int* __restrict__ ei,
                         const float* __restrict__ a_src, const float* __restrict__ a_dst,
                         float* __restrict__ ebuf, unsigned int* __restrict__ m_enc) {
  int t = blockIdx.x * blockDim.x + threadIdx.x;     // over E*HEADS
  if (t >= kE * kHeads) return;
  int e = t >> 2, hd = t & 3;
  int s = ei[e], d = ei[kE + e];
  float v = a_src[s * kHeads + hd] + a_dst[d * kHeads + hd];
  v = (v > 0.f) ? v : kSlope * v;
  ebuf[t] = v;
  atomicMax(&m_enc[d * kHeads + hd], ord_enc(v));
}

// ---------------- edge pass 2: num = exp(e - m[dst]); den[dst] += num ----------------
__global__ void edge_expsum(const int* __restrict__ ei,
                            const unsigned int* __restrict__ m_enc,
                            float* __restrict__ ebuf, float* __restrict__ den) {
  int t = blockIdx.x * blockDim.x + threadIdx.x;     // over E*HEADS
  if (t >= kE * kHeads) return;
  int e = t >> 2, hd = t & 3;
  int d = ei[kE + e];
  float m   = ord_dec(m_enc[d * kHeads + hd]);
  float num = __expf(ebuf[t] - m);
  ebuf[t] = num;                                     // reuse buffer: now holds numerator
  atomicAdd(&den[d * kHeads + hd], num);
}

// ---------------- 1/den (isolated nodes -> 0, never read) ----------------
__global__ void recip_den(float* __restrict__ den) {
  int t = blockIdx.x * blockDim.x + threadIdx.x;     // over N*HEADS
  if (t >= kN * kHeads) return;
  float v = den[t];
  den[t] = (v > 0.f) ? 1.f / v : 0.f;
}

// ---------------- edge pass 3: out1[dst] += alpha * h[src] ----------------
__global__ void edge_message(const int* __restrict__ ei,
                             const float* __restrict__ num,
                             const float* __restrict__ invden,
                             const float* __restrict__ h,
                             float* __restrict__ out1) {
  int t = blockIdx.x * blockDim.x + threadIdx.x;     // over E*HID
  if (t >= kE * kHid) return;
  int e = t >> 7, c = t & 127, hd = c >> 5;
  int s = ei[e], d = ei[kE + e];
  float alpha = num[e * kHeads + hd] * invden[d * kHeads + hd];
  atomicAdd(&out1[(size_t)d * kHid + c], alpha * h[(size_t)s * kHid + c]);
}

// ---------------- add bias + accumulate per-channel sum / sumsq ----------------
__global__ void __launch_bounds__(128)
bias_bnstats(float* __restrict__ buf, const float* __restrict__ bias,
             float* __restrict__ sums, float* __restrict__ sqs, int n) {
  int c = threadIdx.x;                               // 128 channels
  int rows = (n + gridDim.x - 1) / gridDim.x;
  int r0 = blockIdx.x * rows;
  int r1 = min(n, r0 + rows);
  float b = bias[c];
  float s = 0.f, q = 0.f;
  for (int r = r0; r < r1; ++r) {
    float v = buf[(size_t)r * 128 + c] + b;
    buf[(size_t)r * 128 + c] = v;
    s += v; q += v * v;
  }
  atomicAdd(&sums[c], s);
  atomicAdd(&sqs[c], q);
}

// ---------------- BN scale/shift from stats ----------------
__global__ void bn_finalize(const float* __restrict__ sums, const float* __restrict__ sqs,
                            const float* __restrict__ gamma, const float* __restrict__ beta,
                            float* __restrict__ scale, float* __restrict__ shift, float invN) {
  int c = threadIdx.x;
  float mu  = sums[c] * invN;
  float var = sqs[c] * invN - mu * mu;
  float sc  = gamma[c] * rsqrtf(var + kBnEps);
  scale[c] = sc;
  shift[c] = beta[c] - mu * sc;
}

// ---------------- apply BN + ELU in place ----------------
__global__ void bn_apply_elu(float* __restrict__ buf,
                             const float* __restrict__ scale,
                             const float* __restrict__ shift, int total) {
  int t = blockIdx.x * blockDim.x + threadIdx.x;
  if (t >= total) return;
  int c = t & 127;
  float v = buf[t] * scale[c] + shift[c];
  buf[t] = (v > 0.f) ? v : (__expf(v) - 1.f);
}

// ---------------- GIN: z = (1+eps)*out1 ----------------
__global__ void z_init(const float* __restrict__ out1, float* __restrict__ z,
                       const float* __restrict__ epsp, int total) {
  int t = blockIdx.x * blockDim.x + threadIdx.x;
  if (t >= total) return;
  z[t] = (1.f + epsp[0]) * out1[t];
}

// ---------------- GIN aggregate: z[dst] += out1[src] ----------------
__global__ void gin_agg(const int* __restrict__ ei,
                        const float* __restrict__ out1, float* __restrict__ z) {
  int t = blockIdx.x * blockDim.x + threadIdx.x;     // over E*HID
  if (t >= kE * kHid) return;
  int e = t >> 7, c = t & 127;
  int s = ei[e], d = ei[kE + e];
  atomicAdd(&z[(size_t)d * kHid + c], out1[(size_t)s * kHid + c]);
}

// ============================ launch ============================
extern "C" void kernel_launch(void* const* d_in, const int* in_sizes, int n_in,
                              void* d_out, int out_size, void* d_ws, size_t ws_size,
                              hipStream_t stream) {
  (void)in_sizes; (void)n_in; (void)out_size; (void)ws_size;

  const float* x        = (const float*)d_in[0];
  const float* W_gat    = (const float*)d_in[1];
  const float* att_src  = (const float*)d_in[2];
  const float* att_dst  = (const float*)d_in[3];
  const float* bias_gat = (const float*)d_in[4];
  const float* bn1_g    = (const float*)d_in[5];
  const float* bn1_b    = (const float*)d_in[6];
  const float* eps_gin  = (const float*)d_in[7];
  const float* lin_W    = (const float*)d_in[8];
  const float* lin_b    = (const float*)d_in[9];
  const float* bn2_g    = (const float*)d_in[10];
  const float* bn2_b    = (const float*)d_in[11];
  const int*   ei       = (const int*)d_in[12];
  float*       out      = (float*)d_out;

  // ---- workspace layout (floats) ----
  float* ws    = (float*)d_ws;
  float* h     = ws;                                        // N*128
  float* asrc  = h    + (size_t)kN * kHid;                  // N*4
  float* adst  = asrc + (size_t)kN * kHeads;                // N*4
  float* ebuf  = adst + (size_t)kN * kHeads;                // E*4 (e, then num)
  float* out1  = ebuf + (size_t)kE * kHeads;                // N*128   <- zero region start
  unsigned int* menc = (unsigned int*)(out1 + (size_t)kN * kHid); // N*4
  float* den   = (float*)(menc + (size_t)kN * kHeads);      // N*4 (then 1/den)
  float* stats = den  + (size_t)kN * kHeads;                // 4*128: sum1,sq1,sum2,sq2  <- zero end
  float* scsh  = stats + 512;                               // 4*128: scale1,shift1,scale2,shift2
  float* sum1 = stats, *sq1 = stats + 128, *sum2 = stats + 256, *sq2 = stats + 384;
  float* sc1 = scsh, *sh1 = scsh + 128, *sc2 = scsh + 256, *sh2 = scsh + 384;

  // zero: out1 + m_enc + den + stats (one contiguous memset; graph-capture safe)
  size_t zero_bytes = ((size_t)kN * kHid + 2u * (size_t)kN * kHeads + 512u) * sizeof(float);
  hipMemsetAsync(out1, 0, zero_bytes, stream);

  const int B256 = 256;
  const int nNH   = kN * kHeads;                 // 400k
  const int nEH   = kE * kHeads;                 // 6.4M
  const int nNC   = kN * kHid;                   // 12.8M
  const int nEC   = kE * kHid;                   // 204.8M

  // 1) h = x @ W_gat  (fp32 WMMA)
  gemm_f32_wmma<<<(kN / 16 + 3) / 4, 128, 0, stream>>>(x, W_gat, h, kN, kInC, kHid);
  // 2) attention scores
  att_scores<<<(nNH + B256 - 1) / B256, B256, 0, stream>>>(h, att_src, att_dst, asrc, adst);
  // 3) segment max (ordered-uint atomicMax)
  edge_max<<<(nEH + B256 - 1) / B256, B256, 0, stream>>>(ei, asrc, adst, ebuf, menc);
  // 4) exp + segment sum
  edge_expsum<<<(nEH + B256 - 1) / B256, B256, 0, stream>>>(ei, menc, ebuf, den);
  // 5) reciprocal of den
  recip_den<<<(nNH + B256 - 1) / B256, B256, 0, stream>>>(den);
  // 6) weighted message scatter-add (L2-resident: out1/h are 51MB each, L2 is 192MB)
  edge_message<<<(nEC + B256 - 1) / B256, B256, 0, stream>>>(ei, ebuf, den, h, out1);
  // 7) + bias_gat, BN1 stats
  bias_bnstats<<<512, 128, 0, stream>>>(out1, bias_gat, sum1, sq1, kN);
  bn_finalize<<<1, 128, 0, stream>>>(sum1, sq1, bn1_g, bn1_b, sc1, sh1, 1.0f / (float)kN);
  bn_apply_elu<<<(nNC + B256 - 1) / B256, B256, 0, stream>>>(out1, sc1, sh1, nNC);
  // 8) GIN: z = (1+eps)*out1  (reuse h buffer as z)
  z_init<<<(nNC + B256 - 1) / B256, B256, 0, stream>>>(out1, h, eps_gin, nNC);
  // 9) GIN neighbor sum scatter-add
  gin_agg<<<(nEC + B256 - 1) / B256, B256, 0, stream>>>(ei, out1, h);
  // 10) out2 = z @ lin_W  (fp32 WMMA) -> d_out used as fp32 staging
  gemm_f32_wmma<<<(kN / 16 + 3) / 4, 128, 0, stream>>>(h, lin_W, out, kN, kHid, kOutC);
  // 11) + lin_b, BN2 stats, apply BN2+ELU in place on d_out
  bias_bnstats<<<512, 128, 0, stream>>>(out, lin_b, sum2, sq2, kN);
  bn_finalize<<<1, 128, 0, stream>>>(sum2, sq2, bn2_g, bn2_b, sc2, sh2, 1.0f / (float)kN);
  bn_apply_elu<<<(nNC + B256 - 1) / B256, B256, 0, stream>>>(out, sc2, sh2, nNC);
}